// CRF_10797547782364
// MI455X (gfx1250) — compile-verified
//
#include <hip/hip_runtime.h>
#include <hip/hip_bf16.h>
#include <math.h>

// Problem constants (from reference)
#define B_   64
#define L_   8192
#define T_   48
#define CH_  64                 // chunk length along L
#define NC_  (L_ / CH_)         // 128 chunks per batch
#define TP_  50                 // LDS row stride (floats): even -> 8B-aligned b64 frags

typedef float v2f __attribute__((ext_vector_type(2)));
typedef float v8f __attribute__((ext_vector_type(8)));

__device__ __forceinline__ v8f wmma_f32_16x16x4(v2f a, v2f b, v8f c) {
  // 8-arg form: (neg_a, A, neg_b, B, c_mod, C, reuse_a, reuse_b)
  return __builtin_amdgcn_wmma_f32_16x16x4_f32(false, a, false, b, (short)0, c,
                                               false, false);
}

// ---------------------------------------------------------------------------
// Kernel 1: gold-path score (numerator). One block per batch.
// ---------------------------------------------------------------------------
__global__ __launch_bounds__(256) void crf_score_kernel(
    const float* __restrict__ em, const int* __restrict__ tags,
    const float* __restrict__ trans, const float* __restrict__ startT,
    const float* __restrict__ endT, float* __restrict__ scoreOut) {
  const int b = blockIdx.x;
  const int tid = threadIdx.x;
  __shared__ float red[256];

  const int* tg = tags + (size_t)b * L_;
  const float* eb = em + (size_t)b * L_ * T_;

  float s = 0.0f;
  for (int l = tid; l < L_; l += 256) {
    const int cur = tg[l];
    s += eb[(size_t)l * T_ + cur];                 // emission at gold tag
    if (l > 0) s += trans[tg[l - 1] * T_ + cur];   // transition score
  }
  red[tid] = s;
  __syncthreads();
  for (int off = 128; off > 0; off >>= 1) {
    if (tid < off) red[tid] += red[tid + off];
    __syncthreads();
  }
  if (tid == 0) {
    float tot = red[0] + startT[tg[0]] + endT[tg[L_ - 1]];  // mask all-ones
    scoreOut[b] = tot;
  }
}

// ---------------------------------------------------------------------------
// Kernel 2: per-chunk semiring matrix product via f32 WMMA.
// One wave (32 threads) per (batch, chunk). State P (48x48, exp-domain,
// row-normalized) lives in LDS; constant expTrans lives in LDS in
// WMMA-B-fragment order. Per step: P <- rowscale( (P @ expTrans) .* exp(e_l) ).
// ---------------------------------------------------------------------------
__device__ __forceinline__ void renorm_rows(float* P, int lane, float& roA,
                                            float& roB) {
  {
    const int r = lane;  // rows 0..31
    float mx = 0.0f;
    #pragma unroll 8
    for (int j = 0; j < T_; ++j) mx = fmaxf(mx, P[r * TP_ + j]);
    const float inv = 1.0f / mx;
    #pragma unroll 8
    for (int j = 0; j < T_; ++j) P[r * TP_ + j] *= inv;
    roA += __logf(mx);
  }
  if (lane < 16) {
    const int r = lane + 32;  // rows 32..47
    float mx = 0.0f;
    #pragma unroll 8
    for (int j = 0; j < T_; ++j) mx = fmaxf(mx, P[r * TP_ + j]);
    const float inv = 1.0f / mx;
    #pragma unroll 8
    for (int j = 0; j < T_; ++j) P[r * TP_ + j] *= inv;
    roB += __logf(mx);
  }
}

__global__ __launch_bounds__(32) void crf_chunk_kernel(
    const float* __restrict__ em, const float* __restrict__ trans,
    float* __restrict__ Pout, float* __restrict__ roOut) {
  const int gid = blockIdx.x;        // b*NC_ + c
  const int b = gid / NC_;
  const int c = gid % NC_;
  const int lane = threadIdx.x;      // 0..31, full wave
  const int nlo = lane & 15;
  const int khalf = lane >> 4;       // 0 or 1

  __shared__ float P[T_ * TP_];                  // evolving chunk product (exp domain)
  __shared__ float ETF[3 * 12 * 32 * 2];         // expTrans in B-fragment order

  // --- Build constant expTrans B-fragments (each lane fills its own slots) ---
  // B tile (k4, nt): lane holds B[kb][n], B[kb+1][n], kb = 4*k4 + 2*khalf,
  // n = nt*16 + (lane&15)   (mirrors the documented 32-bit A 16x4 layout).
  #pragma unroll
  for (int nt = 0; nt < 3; ++nt) {
    const int n = nt * 16 + nlo;
    #pragma unroll
    for (int k4 = 0; k4 < 12; ++k4) {
      const int kb = k4 * 4 + 2 * khalf;
      const int slot = ((nt * 12 + k4) * 32 + lane) * 2;
      ETF[slot + 0] = __expf(trans[kb * T_ + n]);
      ETF[slot + 1] = __expf(trans[(kb + 1) * T_ + n]);
    }
  }

  const int l0 = (c == 0) ? 1 : c * CH_;   // first step matrix in this chunk
  const int l1 = (c + 1) * CH_;            // exclusive end (<= L_)

  // --- Init P with first matrix: P[i][j] = exp(trans[i][j] + e_{l0}[j]) ---
  {
    const float* e0 = em + ((size_t)b * L_ + l0) * T_;
    for (int idx = lane; idx < T_ * T_; idx += 32) {
      const int i = idx / T_, j = idx % T_;
      P[i * TP_ + j] = __expf(trans[i * T_ + j] + e0[j]);
    }
  }
  __syncthreads();

  float roA = 0.0f, roB = 0.0f;
  renorm_rows(P, lane, roA, roB);

  // --- Fold remaining matrices of the chunk ---
  for (int l = l0 + 1; l < l1; ++l) {
    const float* el = em + ((size_t)b * L_ + l) * T_;
    if (l + 1 < l1) __builtin_prefetch(el + T_, 0, 1);  // global_prefetch_b8

    float w[3];
    #pragma unroll
    for (int nt = 0; nt < 3; ++nt) w[nt] = __expf(el[nt * 16 + nlo]);

    v8f acc[3][3];
    #pragma unroll
    for (int mt = 0; mt < 3; ++mt)
      #pragma unroll
      for (int nt = 0; nt < 3; ++nt) acc[mt][nt] = (v8f)0.0f;

    // 48x48x48 GEMM: D = P @ expTrans, 9 tiles x 12 K-steps of 16x16x4 f32
    #pragma unroll
    for (int mt = 0; mt < 3; ++mt) {
      v2f af[12];
      const int m = mt * 16 + nlo;
      #pragma unroll
      for (int k4 = 0; k4 < 12; ++k4) {
        const int kb = k4 * 4 + 2 * khalf;
        af[k4] = *(const v2f*)&P[m * TP_ + kb];     // ds_load_b64, 8B aligned
      }
      #pragma unroll
      for (int nt = 0; nt < 3; ++nt) {
        #pragma unroll
        for (int k4 = 0; k4 < 12; ++k4) {
          const v2f bf = *(const v2f*)&ETF[((nt * 12 + k4) * 32 + lane) * 2];
          acc[mt][nt] = wmma_f32_16x16x4(af[k4], bf, acc[mt][nt]);
        }
      }
    }

    // All A-fragment reads complete (register-resident accs); overwrite P with
    // the column-scaled result. D layout: vgpr r -> row r + 8*khalf, col nlo.
    #pragma unroll
    for (int mt = 0; mt < 3; ++mt)
      #pragma unroll
      for (int nt = 0; nt < 3; ++nt) {
        const int n = nt * 16 + nlo;
        #pragma unroll
        for (int r = 0; r < 8; ++r) {
          const int m = mt * 16 + r + 8 * khalf;
          P[m * TP_ + n] = acc[mt][nt][r] * w[nt];
        }
      }

    renorm_rows(P, lane, roA, roB);
  }

  // --- Write chunk result: P (exp domain, rowmax=1) + per-row log offsets ---
  float* Pg = Pout + (size_t)gid * (T_ * T_);
  for (int idx = lane; idx < T_ * T_; idx += 32)
    Pg[idx] = P[(idx / T_) * TP_ + (idx % T_)];
  roOut[(size_t)gid * T_ + lane] = roA;                  // rows 0..31
  if (lane < 16) roOut[(size_t)gid * T_ + 32 + lane] = roB;  // rows 32..47
}

// ---------------------------------------------------------------------------
// Kernel 3: sequential fold of chunk matrices into alpha, per batch, then
// partition - score. One block (64 threads) per batch.
// ---------------------------------------------------------------------------
__global__ __launch_bounds__(64) void crf_combine_kernel(
    const float* __restrict__ em, const float* __restrict__ startT,
    const float* __restrict__ endT, const float* __restrict__ Pg,
    const float* __restrict__ rog, const float* __restrict__ scoreIn,
    float* __restrict__ out) {
  const int b = blockIdx.x;
  const int tid = threadIdx.x;

  __shared__ float alpha[T_];
  __shared__ float t[T_];
  __shared__ float wv[T_];
  __shared__ float Pe[T_ * T_];

  if (tid < T_) alpha[tid] = startT[tid] + em[((size_t)b * L_) * T_ + tid];
  __syncthreads();

  for (int c = 0; c < NC_; ++c) {
    const float* Pc = Pg + (size_t)(b * NC_ + c) * (T_ * T_);
    const float* roc = rog + (size_t)(b * NC_ + c) * T_;
    for (int idx = tid; idx < T_ * T_; idx += 64) Pe[idx] = Pc[idx];
    if (tid < T_) t[tid] = alpha[tid] + roc[tid];
    __syncthreads();

    float m = -3.402823466e+38f;
    #pragma unroll 8
    for (int i = 0; i < T_; ++i) m = fmaxf(m, t[i]);
    if (tid < T_) wv[tid] = __expf(t[tid] - m);
    __syncthreads();

    float an = 0.0f;
    if (tid < T_) {
      float s = 0.0f;
      #pragma unroll 8
      for (int i = 0; i < T_; ++i) s += wv[i] * Pe[i * T_ + tid];
      an = m + __logf(s);
    }
    __syncthreads();
    if (tid < T_) alpha[tid] = an;
    __syncthreads();
  }

  if (tid == 0) {
    float m = -3.402823466e+38f;
    for (int j = 0; j < T_; ++j) m = fmaxf(m, alpha[j] + endT[j]);
    float s = 0.0f;
    for (int j = 0; j < T_; ++j) s += __expf(alpha[j] + endT[j] - m);
    const float partition = m + __logf(s);
    out[b] = partition - scoreIn[b];
  }
}

// ---------------------------------------------------------------------------
// Launch
// ---------------------------------------------------------------------------
extern "C" void kernel_launch(void* const* d_in, const int* in_sizes, int n_in,
                              void* d_out, int out_size, void* d_ws,
                              size_t ws_size, hipStream_t stream) {
  const float* em     = (const float*)d_in[0];
  const int*   tags   = (const int*)d_in[1];
  // d_in[2] = mask (all ones in this problem; unused)
  const float* trans  = (const float*)d_in[3];
  const float* startT = (const float*)d_in[4];
  const float* endT   = (const float*)d_in[5];
  float* out = (float*)d_out;

  // Workspace layout
  char* ws = (char*)d_ws;
  float* scoreBuf = (float*)ws;                                   // B floats
  float* Pbuf     = (float*)(ws + 256);                           // B*NC*T*T
  float* roBuf    = (float*)(ws + 256 + (size_t)B_ * NC_ * T_ * T_ * 4);  // B*NC*T
  (void)ws_size; (void)in_sizes; (void)n_in; (void)out_size;

  crf_score_kernel<<<B_, 256, 0, stream>>>(em, tags, trans, startT, endT,
                                           scoreBuf);
  crf_chunk_kernel<<<B_ * NC_, 32, 0, stream>>>(em, trans, Pbuf, roBuf);
  crf_combine_kernel<<<B_, 64, 0, stream>>>(em, startT, endT, Pbuf, roBuf,
                                            scoreBuf, out);
}